// SASABlock_47519518163543
// MI455X (gfx1250) — compile-verified
//
#include <hip/hip_runtime.h>
#include <hip/hip_bf16.h>
#include <math.h>

#define C_CH 128
#define Hh   112
#define HW   12544      // 112*112
#define NPIX 1605632    // 128*12544
#define HPAD 118
#define HWP  13924      // 118*118
#define PADN 1782272    // 128*13924
#define PCHUNK 98       // 12544 = 128 * 98

typedef __attribute__((ext_vector_type(2))) float v2f;
typedef __attribute__((ext_vector_type(8))) float v8f;

#if __has_builtin(__builtin_amdgcn_wmma_f32_16x16x4_f32)
#define USE_WMMA_F32 1
#else
#define USE_WMMA_F32 0
#endif

// ---------------------------------------------------------------------------
// NCHW -> NHWC tiled transpose for the input tensor (LDS 32x33 tile).
// ---------------------------------------------------------------------------
__global__ __launch_bounds__(256) void transpose_c2p(
    const float* __restrict__ src /*[128,HW]*/, float* __restrict__ dst /*[HW,128]*/)
{
  __shared__ float tile[32][33];
  const int tx = threadIdx.x;       // 0..31
  const int ty = threadIdx.y;       // 0..7
  const int p0 = blockIdx.x * 32;
  const int c0 = blockIdx.y * 32;
  #pragma unroll
  for (int k = 0; k < 32; k += 8)
    tile[ty + k][tx] = src[(size_t)(c0 + ty + k) * HW + p0 + tx];
  __syncthreads();
  #pragma unroll
  for (int k = 0; k < 32; k += 8)
    dst[(size_t)(p0 + ty + k) * C_CH + c0 + tx] = tile[tx][ty + k];
}

// ---------------------------------------------------------------------------
// GEMM (NHWC): Y[n,m] = sum_k Wm[m,k] * X[n,k] + bias[m]
// Wm: [128x128] row-major (out,in). X: [N,128]. Y: [N,128].
// One wave = one 16(M)x16(N) tile via V_WMMA_F32_16X16X4_F32, K fully unrolled.
// A frag: lane l holds W row m0+l, VGPR pair = K {+0,+1} (lanes<16) / {+2,+3}.
// B frag: lane l holds X column n0+l, same K split -> contiguous b64 loads.
// D: VGPR v = row m0+v (+8 for lanes>=16), col = lane -> 8 contiguous floats.
// ---------------------------------------------------------------------------
template<int N>
__global__ __launch_bounds__(128) void gemm_nhwc(
    const float* __restrict__ Wm, const float* __restrict__ X,
    const float* __restrict__ bias, float* __restrict__ Y)
{
  const int wave = threadIdx.x >> 5;
  const int lane = threadIdx.x & 31;
  const int half = lane >> 4;
  const int l    = lane & 15;
  const int m0   = blockIdx.y << 4;
  const int n0   = (blockIdx.x * 4 + wave) << 4;
  if (n0 >= N) return;              // wave-uniform; EXEC stays all-ones

  const int n  = n0 + l;
  const int nc = (n < N) ? n : (N - 1);          // clamp tail lanes (masked at store)
  const float* __restrict__ xr = X  + (size_t)nc * C_CH + 2 * half;
  const float* __restrict__ wr = Wm + (size_t)(m0 + l) * C_CH + 2 * half;

#if USE_WMMA_F32
  v8f acc = {};
  #pragma unroll
  for (int k = 0; k < C_CH; k += 4) {
    v2f a, b;
    a.x = wr[k]; a.y = wr[k + 1];   // b64, immediate offset 4k
    b.x = xr[k]; b.y = xr[k + 1];   // b64, immediate offset 4k
    acc = __builtin_amdgcn_wmma_f32_16x16x4_f32(
        false, a, false, b, (short)0, acc, false, false);
  }
  if (n < N) {
    float* yo = Y + (size_t)n * C_CH + m0 + 8 * half;
    #pragma unroll
    for (int v = 0; v < 8; ++v)
      yo[v] = acc[v] + (bias ? bias[m0 + 8 * half + v] : 0.f);
  }
#else
  float acc[8] = {};
  for (int k = 0; k < C_CH; ++k) {
    const float xv = X[(size_t)nc * C_CH + k];
    #pragma unroll
    for (int v = 0; v < 8; ++v)
      acc[v] = fmaf(Wm[(size_t)(m0 + 8 * half + v) * C_CH + k], xv, acc[v]);
  }
  if (n < N) {
    float* yo = Y + (size_t)n * C_CH + m0 + 8 * half;
    #pragma unroll
    for (int v = 0; v < 8; ++v)
      yo[v] = acc[v] + (bias ? bias[m0 + 8 * half + v] : 0.f);
  }
#endif
}

// Zero-pad (3 each side) NHWC activation: [HW,128] -> [HWP,128].
__global__ __launch_bounds__(256) void pad_nhwc(
    const float* __restrict__ src, float* __restrict__ dst)
{
  const int idx = blockIdx.x * 256 + threadIdx.x;   // PADN threads exactly
  const int pp = idx >> 7, c = idx & 127;
  const int xp = pp / HPAD, yp = pp - xp * HPAD;
  const int sx = xp - 3, sy = yp - 3;
  const bool in = ((unsigned)sx < (unsigned)Hh) && ((unsigned)sy < (unsigned)Hh);
  dst[idx] = in ? src[(size_t)(sx * Hh + sy) * C_CH + c] : 0.f;
}

// Deterministic two-pass BN stats over NHWC [HW,128]; biased variance.
__global__ __launch_bounds__(128) void bnstats_partial(
    const float* __restrict__ X, float* __restrict__ ps, float* __restrict__ psq)
{
  const int c = threadIdx.x;
  const int p0 = blockIdx.x * PCHUNK;
  float s = 0.f, sq = 0.f;
  for (int i = 0; i < PCHUNK; ++i) {
    const float v = X[(size_t)(p0 + i) * C_CH + c];
    s += v; sq += v * v;
  }
  ps [blockIdx.x * C_CH + c] = s;
  psq[blockIdx.x * C_CH + c] = sq;
}

__global__ __launch_bounds__(128) void bnstats_final(
    const float* __restrict__ ps, const float* __restrict__ psq,
    const float* __restrict__ g, const float* __restrict__ b,
    float* __restrict__ scale, float* __restrict__ shift)
{
  const int c = threadIdx.x;
  float s = 0.f, sq = 0.f;
  for (int i = 0; i < 128; ++i) { s += ps[i * C_CH + c]; sq += psq[i * C_CH + c]; }
  const float mean = s * (1.f / (float)HW);
  const float var  = sq * (1.f / (float)HW) - mean * mean;
  const float sc   = g[c] * rsqrtf(var + 1e-3f);
  scale[c] = sc;
  shift[c] = b[c] - mean * sc;
}

__global__ __launch_bounds__(256) void bnapply_relu(
    const float* __restrict__ Yin, const float* __restrict__ scale,
    const float* __restrict__ shift, float* __restrict__ Out)
{
  const int idx = blockIdx.x * 256 + threadIdx.x;   // NPIX exactly
  const int c = idx & 127;
  Out[idx] = fmaxf(fmaf(Yin[idx], scale[c], shift[c]), 0.f);
}

// bias49[i*7+j] = sum_c rel_h[c,i] + sum_c rel_w[c,j] (no-shared-index einsum)
__global__ void relbias_kernel(const float* __restrict__ rel_h,
                               const float* __restrict__ rel_w,
                               float* __restrict__ bias49)
{
  const int s = threadIdx.x;
  if (s < 49) {
    const int i = s / 7, j = s - i * 7;
    float a = 0.f, b = 0.f;
    for (int t = 0; t < 64; ++t) { a += rel_h[t * 7 + i]; b += rel_w[t * 7 + j]; }
    bias49[s] = a + b;
  }
}

// SASA core after the flat-index relabel. Block = one (row x, col-block y0)
// group; thread = channel c (128 lanes -> every global access is a contiguous
// 512B block-wide transaction). Q/K/V/S are NHWC.
__global__ __launch_bounds__(128) void sasa_attn(
    const float* __restrict__ Q, const float* __restrict__ Kp,
    const float* __restrict__ Vp, const float* __restrict__ bias49,
    float* __restrict__ S)
{
  __shared__ float sb[49];
  if (threadIdx.x < 49) sb[threadIdx.x] = bias49[threadIdx.x];
  __syncthreads();

  const int g  = blockIdx.x;          // 784 groups = 112 rows * 7 col-blocks
  const int c  = threadIdx.x;
  const int x  = g / 7;
  const int y0 = (g - x * 7) * 16;

  const float* qb = Q  + (size_t)(x * Hh   + y0) * C_CH + c;
  const float* kb = Kp + (size_t)(x * HPAD + y0) * C_CH + c;
  const float* vb = Vp + (size_t)(x * HPAD + y0) * C_CH + c;

  float qv[16], qsum = 0.f;
  #pragma unroll
  for (int d = 0; d < 16; ++d) { qv[d] = qb[(size_t)d * C_CH]; qsum += qv[d]; }

  float qk[49];
  #pragma unroll
  for (int i = 0; i < 7; ++i) {
    float row[22];
    #pragma unroll
    for (int u = 0; u < 22; ++u) row[u] = kb[(size_t)(i * HPAD + u) * C_CH];
    #pragma unroll
    for (int j = 0; j < 7; ++j) {
      float a = 0.f;
      #pragma unroll
      for (int d = 0; d < 16; ++d) a = fmaf(qv[d], row[d + j], a);
      qk[i * 7 + j] = a;
    }
  }

  float mx = -3.4e38f;
  #pragma unroll
  for (int s = 0; s < 49; ++s) { qk[s] = fmaf(qsum, sb[s], qk[s]); mx = fmaxf(mx, qk[s]); }
  float sum = 0.f;
  #pragma unroll
  for (int s = 0; s < 49; ++s) { qk[s] = __expf(qk[s] - mx); sum += qk[s]; }
  const float inv = 1.f / sum;

  float o[16];
  #pragma unroll
  for (int d = 0; d < 16; ++d) o[d] = 0.f;
  #pragma unroll
  for (int i = 0; i < 7; ++i) {
    float row[22];
    #pragma unroll
    for (int u = 0; u < 22; ++u) row[u] = vb[(size_t)(i * HPAD + u) * C_CH];
    #pragma unroll
    for (int j = 0; j < 7; ++j) {
      const float wv = qk[i * 7 + j] * inv;
      #pragma unroll
      for (int d = 0; d < 16; ++d) o[d] = fmaf(wv, row[d + j], o[d]);
    }
  }
  float* so = S + (size_t)(x * Hh + y0) * C_CH + c;
  #pragma unroll
  for (int d = 0; d < 16; ++d) so[(size_t)d * C_CH] = o[d];
}

// out (NCHW) = relu(x + bn2(y2)); y2 is NHWC.
__global__ __launch_bounds__(256) void residual_relu(
    const float* __restrict__ x, const float* __restrict__ y2,
    const float* __restrict__ scale, const float* __restrict__ shift,
    float* __restrict__ out)
{
  const int idx = blockIdx.x * 256 + threadIdx.x;   // NPIX exactly
  const int c = idx / HW, p = idx - c * HW;
  const float v = y2[(size_t)p * C_CH + c];
  out[idx] = fmaxf(x[idx] + fmaf(v, scale[c], shift[c]), 0.f);
}

extern "C" void kernel_launch(void* const* d_in, const int* in_sizes, int n_in,
                              void* d_out, int out_size, void* d_ws, size_t ws_size,
                              hipStream_t stream)
{
  (void)in_sizes; (void)n_in; (void)out_size; (void)ws_size;
  const float* x    = (const float*)d_in[0];
  const float* w1   = (const float*)d_in[1];
  const float* g1   = (const float*)d_in[2];
  const float* b1   = (const float*)d_in[3];
  const float* qw   = (const float*)d_in[4];
  const float* qb   = (const float*)d_in[5];
  const float* kw   = (const float*)d_in[6];
  const float* kb   = (const float*)d_in[7];
  const float* vw   = (const float*)d_in[8];
  const float* vb   = (const float*)d_in[9];
  const float* relh = (const float*)d_in[10];
  const float* relw = (const float*)d_in[11];
  const float* ga   = (const float*)d_in[12];
  const float* ba   = (const float*)d_in[13];
  const float* w2   = (const float*)d_in[14];
  const float* g2   = (const float*)d_in[15];
  const float* b2   = (const float*)d_in[16];

  float* ws = (float*)d_ws;
  float* xT_out1 = ws;                         // xT, later out1      [HW,128]
  float* y1S     = ws + (size_t)NPIX;          // y1, later S         [HW,128]
  float* Qy2     = ws + (size_t)2 * NPIX;      // Q,  later y2        [HW,128]
  float* o1pA2   = ws + (size_t)3 * NPIX;      // out1p, later A2     [HWP,128]
  float* Kp      = o1pA2 + (size_t)PADN;       //                     [HWP,128]
  float* Vp      = Kp + (size_t)PADN;          //                     [HWP,128]
  float* small   = Vp + (size_t)PADN;
  float* ps      = small;                      // 128*128
  float* psq     = small + 16384;              // 128*128
  float* scale1  = small + 32768;  float* shift1 = scale1 + 128;
  float* scaleA  = scale1 + 256;   float* shiftA = scale1 + 384;
  float* scale2  = scale1 + 512;   float* shift2 = scale1 + 640;
  float* dbias49 = scale1 + 768;

  const dim3 gN (196, 8);   // N=12544: 784 tiles exactly
  const dim3 gNp(218, 8);   // N=13924: 871 tiles (last partial; clamped+masked)
  const dim3 gT (392, 4), bT(32, 8);

  relbias_kernel<<<1, 64, 0, stream>>>(relh, relw, dbias49);
  transpose_c2p<<<gT, bT, 0, stream>>>(x, xT_out1);                    // x -> NHWC
  // block1: conv1x1 -> BN -> ReLU
  gemm_nhwc<HW><<<gN, 128, 0, stream>>>(w1, xT_out1, nullptr, y1S);
  bnstats_partial<<<128, 128, 0, stream>>>(y1S, ps, psq);
  bnstats_final<<<1, 128, 0, stream>>>(ps, psq, g1, b1, scale1, shift1);
  bnapply_relu<<<6272, 256, 0, stream>>>(y1S, scale1, shift1, xT_out1); // out1
  // q/k/v 1x1 convs; k,v on explicitly padded 118x118 grid
  pad_nhwc<<<6962, 256, 0, stream>>>(xT_out1, o1pA2);
  gemm_nhwc<HW ><<<gN,  128, 0, stream>>>(qw, xT_out1, qb, Qy2);
  gemm_nhwc<HWP><<<gNp, 128, 0, stream>>>(kw, o1pA2,  kb, Kp);
  gemm_nhwc<HWP><<<gNp, 128, 0, stream>>>(vw, o1pA2,  vb, Vp);
  // SASA core (post-relabel algebra)
  sasa_attn<<<784, 128, 0, stream>>>(Qy2, Kp, Vp, dbias49, y1S);        // S
  // norm1 + act
  bnstats_partial<<<128, 128, 0, stream>>>(y1S, ps, psq);
  bnstats_final<<<1, 128, 0, stream>>>(ps, psq, ga, ba, scaleA, shiftA);
  bnapply_relu<<<6272, 256, 0, stream>>>(y1S, scaleA, shiftA, o1pA2);   // A2
  // conv (no bias) + norm2 + residual + act
  gemm_nhwc<HW><<<gN, 128, 0, stream>>>(w2, o1pA2, nullptr, Qy2);       // y2
  bnstats_partial<<<128, 128, 0, stream>>>(Qy2, ps, psq);
  bnstats_final<<<1, 128, 0, stream>>>(ps, psq, g2, b2, scale2, shift2);
  residual_relu<<<6272, 256, 0, stream>>>(x, Qy2, scale2, shift2, (float*)d_out);
}